// RecurrentGCN_50465865728448
// MI455X (gfx1250) — compile-verified
//
#include <hip/hip_runtime.h>

// DCRNN cell (K=1) fused kernel for gfx1250 / MI455X.
//
// Roofline: 16.6 GFLOP vs ~390 MB HBM traffic (x dominates: 307 MB).
// HBM floor ~17us @ 23.3 TB/s; fp32 WMMA (V_WMMA_F32_16X16X4_F32) keeps the
// reference's fp32 precision and its throughput is within ~1.5x of that
// floor, so fp32 matrix cores are the right precision/throughput point.
//
// Weights (3 x 288x32, combined fwd+bwd) are summed into LDS once per block,
// pre-packed so each WMMA B fragment is one ds_load_b64:
//   B 4x16 layout: lanes 0-15 hold {K0,K1} (VGPR0,VGPR1), lanes 16-31 {K2,K3},
//   lane%16 = N  (mirrors the documented f32 A 16x4 layout).
// A fragments load directly from global: per-lane GLOBAL_LOAD_B64 at
//   row = base + lane%16, col = 4k + 2*(lane>=16)  -> (VGPR0,VGPR1) = (K0/K2, K1/K3).

#define N_NODES 300000
#define F_IN    256
#define HID     32
#define CIN     288
#define OUT_DIM 3

typedef float v2f __attribute__((ext_vector_type(2)));
typedef float v8f __attribute__((ext_vector_type(8)));

#define MB    4                        // 16-row tiles per wave
#define WAVES 8                        // waves per 256-thread block
#define ROWS_PER_BLOCK (WAVES * MB * 16)   // 512

// ---- LDS layout (float indices) ----
#define OFF_WZ   0                     // 9216 packed
#define OFF_WR   9216
#define OFF_WH   18432
#define OFF_BZ   27648
#define OFF_BR   27680
#define OFF_BH   27712
#define OFF_WL   27744                 // 32x3 row-major
#define OFF_BL   27840                 // 3 (+1 pad)
#define OFF_STAGE 27844
#define STAGE_STRIDE 34                // 16x34 per wave (padded vs 64 banks)
#define STAGE_PER_WAVE (16 * STAGE_STRIDE)
#define LDS_FLOATS (OFF_STAGE + WAVES * STAGE_PER_WAVE)   // 32196 -> 128784 B

__device__ __forceinline__ float sigmoidf_fast(float v) {
    return 1.0f / (1.0f + __expf(-v));
}
__device__ __forceinline__ float tanhf_fast(float v) {
    float e = __expf(-2.0f * v);
    return (1.0f - e) / (1.0f + e);
}

__launch_bounds__(256, 1)
__global__ void dcrnn_fused_kernel(const float* __restrict__ x,
                                   const float* __restrict__ h,
                                   const float* __restrict__ Wz,
                                   const float* __restrict__ bz,
                                   const float* __restrict__ Wr,
                                   const float* __restrict__ br,
                                   const float* __restrict__ Wh,
                                   const float* __restrict__ bh,
                                   const float* __restrict__ Wl,
                                   const float* __restrict__ bl,
                                   float* __restrict__ out,
                                   float* __restrict__ hn_out) {
    extern __shared__ float lds[];
    const int tid = threadIdx.x;

    // ---- stage combined weights into LDS, packed for ds_load_b64 B frags ----
    // pos(K,c) = (chunk*2 + ((K>>1)&1)) * 64 + c*2 + (K&1), chunk = K/4
    for (int i = tid; i < 9216; i += 256) {
        const int K = i >> 5, c = i & 31;
        const int pos = (((K >> 2) << 1) + ((K >> 1) & 1)) * 64 + (c << 1) + (K & 1);
        lds[OFF_WZ + pos] = Wz[i] + Wz[9216 + i];
        lds[OFF_WR + pos] = Wr[i] + Wr[9216 + i];
        lds[OFF_WH + pos] = Wh[i] + Wh[9216 + i];
    }
    if (tid < 32) {
        lds[OFF_BZ + tid] = bz[tid];
        lds[OFF_BR + tid] = br[tid];
        lds[OFF_BH + tid] = bh[tid];
    }
    if (tid < 96) lds[OFF_WL + tid] = Wl[tid];
    if (tid < 3)  lds[OFF_BL + tid] = bl[tid];
    __syncthreads();

    const int lane = tid & 31;
    const int wid  = tid >> 5;
    const int nl   = lane & 15;        // A: row M; B/C/D: column N (mod 16)
    const int hi   = lane >> 4;        // lane half selects K pair {0,1} vs {2,3}
    const int off  = hi << 1;

    float* stage = &lds[OFF_STAGE + wid * STAGE_PER_WAVE];

    const int rbase = blockIdx.x * ROWS_PER_BLOCK + wid * (MB * 16);

    int r0[MB];
    const float* px[MB];
    const float* ph[MB];
#pragma unroll
    for (int t = 0; t < MB; ++t) {
        r0[t] = rbase + t * 16;
        int rm = r0[t] + nl;
        if (rm > N_NODES - 1) rm = N_NODES - 1;   // clamp reads; stores predicated
        px[t] = x + (size_t)rm * F_IN + off;
        ph[t] = h + (size_t)rm * HID + off;
    }

    v8f accZ[MB][2], accR[MB][2], accP[MB][2];
#pragma unroll
    for (int t = 0; t < MB; ++t)
#pragma unroll
        for (int c = 0; c < 2; ++c) {
            accZ[t][c] = v8f{0.f,0.f,0.f,0.f,0.f,0.f,0.f,0.f};
            accR[t][c] = v8f{0.f,0.f,0.f,0.f,0.f,0.f,0.f,0.f};
            accP[t][c] = v8f{0.f,0.f,0.f,0.f,0.f,0.f,0.f,0.f};
        }

    // Per-lane constant base into packed weight blocks.
    const float* bz0 = &lds[OFF_WZ + hi * 64 + nl * 2];
    const float* br0 = &lds[OFF_WR + hi * 64 + nl * 2];
    const float* bh0 = &lds[OFF_WH + hi * 64 + nl * 2];

    // ---- main K loop: x part (K = 0..255), feeds Z, R and P(=x@Whx) ----
#pragma unroll 2
    for (int k = 0; k < 64; ++k) {
        v2f a[MB];
#pragma unroll
        for (int t = 0; t < MB; ++t) a[t] = *(const v2f*)(px[t] + 4 * k);
#pragma unroll
        for (int ct = 0; ct < 2; ++ct) {
            const v2f bzf = *(const v2f*)(bz0 + k * 128 + ct * 32);
            const v2f brf = *(const v2f*)(br0 + k * 128 + ct * 32);
            const v2f bhf = *(const v2f*)(bh0 + k * 128 + ct * 32);
#pragma unroll
            for (int t = 0; t < MB; ++t) {
                accZ[t][ct] = __builtin_amdgcn_wmma_f32_16x16x4_f32(
                    false, a[t], false, bzf, (short)0, accZ[t][ct], false, false);
                accR[t][ct] = __builtin_amdgcn_wmma_f32_16x16x4_f32(
                    false, a[t], false, brf, (short)0, accR[t][ct], false, false);
                accP[t][ct] = __builtin_amdgcn_wmma_f32_16x16x4_f32(
                    false, a[t], false, bhf, (short)0, accP[t][ct], false, false);
            }
        }
    }

    // ---- K loop: h part (K = 256..287), feeds Z and R only ----
#pragma unroll
    for (int k = 0; k < 8; ++k) {
        v2f a[MB];
#pragma unroll
        for (int t = 0; t < MB; ++t) a[t] = *(const v2f*)(ph[t] + 4 * k);
#pragma unroll
        for (int ct = 0; ct < 2; ++ct) {
            const v2f bzf = *(const v2f*)(bz0 + (64 + k) * 128 + ct * 32);
            const v2f brf = *(const v2f*)(br0 + (64 + k) * 128 + ct * 32);
#pragma unroll
            for (int t = 0; t < MB; ++t) {
                accZ[t][ct] = __builtin_amdgcn_wmma_f32_16x16x4_f32(
                    false, a[t], false, bzf, (short)0, accZ[t][ct], false, false);
                accR[t][ct] = __builtin_amdgcn_wmma_f32_16x16x4_f32(
                    false, a[t], false, brf, (short)0, accR[t][ct], false, false);
            }
        }
    }

    // ---- epilogue per row-tile ----
    for (int t = 0; t < MB; ++t) {
        // R = sigmoid(accR + br); stage R*h for the K=32 correction GEMM.
        float hval[2][8];
#pragma unroll
        for (int ct = 0; ct < 2; ++ct) {
            const int n = ct * 16 + nl;
            const float brv = lds[OFF_BR + n];
#pragma unroll
            for (int v = 0; v < 8; ++v) {
                const int m = v + 8 * hi;
                int row = r0[t] + m;
                if (row > N_NODES - 1) row = N_NODES - 1;
                const float hv = h[(size_t)row * HID + n];
                hval[ct][v] = hv;
                const float r = sigmoidf_fast(accR[t][ct][v] + brv);
                stage[m * STAGE_STRIDE + n] = r * hv;
            }
        }
        __syncthreads();

        // accP += (R*h) @ Whh  (Wh chunks 64..71)
#pragma unroll
        for (int kk = 0; kk < 8; ++kk) {
            const v2f a = *(const v2f*)(stage + nl * STAGE_STRIDE + 4 * kk + off);
#pragma unroll
            for (int ct = 0; ct < 2; ++ct) {
                const v2f bf = *(const v2f*)(bh0 + (64 + kk) * 128 + ct * 32);
                accP[t][ct] = __builtin_amdgcn_wmma_f32_16x16x4_f32(
                    false, a, false, bf, (short)0, accP[t][ct], false, false);
            }
        }

        // Hn = Z*h + (1-Z)*tanh(P + bh); store Hn; stage relu(Hn) for readout.
#pragma unroll
        for (int ct = 0; ct < 2; ++ct) {
            const int n = ct * 16 + nl;
            const float bzv = lds[OFF_BZ + n];
            const float bhv = lds[OFF_BH + n];
#pragma unroll
            for (int v = 0; v < 8; ++v) {
                const int m = v + 8 * hi;
                const int row = r0[t] + m;
                const float z  = sigmoidf_fast(accZ[t][ct][v] + bzv);
                const float ht = tanhf_fast(accP[t][ct][v] + bhv);
                const float hn = z * hval[ct][v] + (1.0f - z) * ht;
                if (row < N_NODES) hn_out[(size_t)row * HID + n] = hn;
                stage[m * STAGE_STRIDE + n] = hn > 0.0f ? hn : 0.0f;
            }
        }
        __syncthreads();

        // readout: out = relu(Hn) @ Wl + bl   (32 -> 3, tiny)
        if (lane < 16) {
            const int row = r0[t] + lane;
            if (row < N_NODES) {
                float s0 = lds[OFF_BL + 0];
                float s1 = lds[OFF_BL + 1];
                float s2 = lds[OFF_BL + 2];
#pragma unroll
                for (int c = 0; c < HID; ++c) {
                    const float hv = stage[lane * STAGE_STRIDE + c];
                    s0 += hv * lds[OFF_WL + c * 3 + 0];
                    s1 += hv * lds[OFF_WL + c * 3 + 1];
                    s2 += hv * lds[OFF_WL + c * 3 + 2];
                }
                out[(size_t)row * 3 + 0] = s0;
                out[(size_t)row * 3 + 1] = s1;
                out[(size_t)row * 3 + 2] = s2;
            }
        }
    }
}

extern "C" void kernel_launch(void* const* d_in, const int* in_sizes, int n_in,
                              void* d_out, int out_size, void* d_ws, size_t ws_size,
                              hipStream_t stream) {
    (void)in_sizes; (void)n_in; (void)out_size; (void)d_ws; (void)ws_size;

    const float* x  = (const float*)d_in[0];
    // d_in[1] = edge_index, d_in[2] = edge_weight : unused (K == 1 path)
    const float* h  = (const float*)d_in[3];
    const float* Wz = (const float*)d_in[4];
    const float* bz = (const float*)d_in[5];
    const float* Wr = (const float*)d_in[6];
    const float* br = (const float*)d_in[7];
    const float* Wh = (const float*)d_in[8];
    const float* bh = (const float*)d_in[9];
    const float* Wl = (const float*)d_in[10];
    const float* bl = (const float*)d_in[11];

    float* out = (float*)d_out;                       // [N,3] first
    float* hn  = out + (size_t)N_NODES * OUT_DIM;     // then [N,32]

    const int blocks = (N_NODES + ROWS_PER_BLOCK - 1) / ROWS_PER_BLOCK;  // 586
    const size_t shmem = (size_t)LDS_FLOATS * sizeof(float);             // ~126 KB

    (void)hipFuncSetAttribute((const void*)dcrnn_fused_kernel,
                              hipFuncAttributeMaxDynamicSharedMemorySize,
                              (int)shmem);

    hipLaunchKernelGGL(dcrnn_fused_kernel, dim3(blocks), dim3(256), shmem, stream,
                       x, h, Wz, bz, Wr, br, Wh, bh, Wl, bl, out, hn);
}